// RoI_18777597018773
// MI455X (gfx1250) — compile-verified
//
#include <hip/hip_runtime.h>
#include <hip/hip_bf16.h>

// ---------------------------------------------------------------------------
// Faster-RCNN RoI head on MI455X (gfx1250, wave32).
//   * GEMMs: V_WMMA_F32_16X16X4_F32 (fp32 tensor path == reference precision).
//     FC6 dominates (102.8 GFLOP fp32); box_features (200MB) streams once
//     (~8.6us @ 23.3 TB/s), W6 (51MB) stays resident in the 192MB L2.
//     Double-buffered LDS pipeline, B tile stored transposed [N][K+2] so every
//     WMMA B-fragment is a single aligned ds_load_b64 (no v_mov repacking).
//   * Sort: exact per-image descending bitonic (tie-break by index, matches
//     jax.lax.top_k) in 28 launches (LDS-resident phases for jj<=1024).
//   * NMS: mask-matrix form; a single wave replays the exact greedy scan.
// ---------------------------------------------------------------------------

typedef float v2f __attribute__((ext_vector_type(2)));
typedef float v8f __attribute__((ext_vector_type(8)));

#define BIMG 4
#define RPROP 1000
#define NCLS 91
#define POOLD 12544
#define REPD 1024
#define NROWS (BIMG * RPROP)          // 4000
#define NCAND (RPROP * (NCLS - 1))    // 90000 candidates / image
#define NSORT 131072                  // 2^17 padded sort size / image
#define NSORT_LOG 17
#define KEEPN 1024                    // NMS slots (first 1000 = PRE_NMS_K)
#define BBOX_CLIP 4.135166556742356f  // log(1000/16)

// ---------------- WMMA fp32 GEMM:  C = relu?(A[M,K] @ W[K,N] + bias) -------
#define MT 64
#define NT 128
#define KTL 32                        // K is a multiple of 32 for all calls
#define LDP (KTL + 2)                 // LDS row pitch: even -> 8B-aligned v2f

__global__ __launch_bounds__(256) void gemm_wmma_f32(
    const float* __restrict__ A, const float* __restrict__ W,
    const float* __restrict__ bias, float* __restrict__ C,
    int M, int N, int K, int ldc, int relu)
{
  __shared__ float As[2][MT][LDP];    // A tile, [m][k]
  __shared__ float Bs[2][NT][LDP];    // B tile TRANSPOSED, [n][k]

  const int tid  = threadIdx.x;
  const int wave = tid >> 5;
  const int lane = tid & 31;
  const int lm   = lane & 15;
  const int hi   = lane >> 4;
  const int m0   = (wave & 3) << 4;
  const int n0   = (wave >> 2) << 6;
  const int blockM = blockIdx.y * MT;
  const int blockN = blockIdx.x * NT;

  v8f acc[4] = {};                    // 16x64 per wave
  float4 aReg[2];
  float4 bReg[4];

#define LOAD_TILE(KT_)                                                     \
  {                                                                        \
    int kt_ = (KT_);                                                       \
    _Pragma("unroll")                                                      \
    for (int u = 0; u < 2; ++u) {                                          \
      int q = tid + (u << 8);                                              \
      int ar = q >> 3, kq = (q & 7) << 2;                                  \
      int gm = blockM + ar;                                                \
      float4 v = make_float4(0.f, 0.f, 0.f, 0.f);                          \
      if (gm < M) {                                                        \
        const float* p = A + (size_t)gm * K + kt_ + kq;                    \
        v = *(const float4*)p;                                             \
        if (kt_ + 2 * KTL < K) __builtin_prefetch(p + 2 * KTL, 0, 0);      \
      }                                                                    \
      aReg[u] = v;                                                         \
    }                                                                      \
    _Pragma("unroll")                                                      \
    for (int u = 0; u < 4; ++u) {                                          \
      int q = tid + (u << 8);                                              \
      int br = q >> 5, nc4 = (q & 31) << 2;                                \
      int gn = blockN + nc4;                                               \
      float4 v = make_float4(0.f, 0.f, 0.f, 0.f);                          \
      const float* wp = W + (size_t)(kt_ + br) * N + gn;                   \
      if (((N & 3) == 0) && (gn + 3 < N)) {                                \
        v = *(const float4*)wp;       /* aligned: N%4==0 */                \
      } else {                                                             \
        if (gn + 0 < N) v.x = wp[0];                                       \
        if (gn + 1 < N) v.y = wp[1];                                       \
        if (gn + 2 < N) v.z = wp[2];                                       \
        if (gn + 3 < N) v.w = wp[3];                                       \
      }                                                                    \
      bReg[u] = v;                                                         \
    }                                                                      \
  }

#define STORE_TILE(BUF_)                                                   \
  {                                                                        \
    _Pragma("unroll")                                                      \
    for (int u = 0; u < 2; ++u) {                                          \
      int q = tid + (u << 8);                                              \
      int ar = q >> 3, kq = (q & 7) << 2;                                  \
      As[BUF_][ar][kq + 0] = aReg[u].x; As[BUF_][ar][kq + 1] = aReg[u].y;  \
      As[BUF_][ar][kq + 2] = aReg[u].z; As[BUF_][ar][kq + 3] = aReg[u].w;  \
    }                                                                      \
    _Pragma("unroll")                                                      \
    for (int u = 0; u < 4; ++u) {                                          \
      int q = tid + (u << 8);                                              \
      int br = q >> 5, nc4 = (q & 31) << 2;                                \
      Bs[BUF_][nc4 + 0][br] = bReg[u].x;                                   \
      Bs[BUF_][nc4 + 1][br] = bReg[u].y;                                   \
      Bs[BUF_][nc4 + 2][br] = bReg[u].z;                                   \
      Bs[BUF_][nc4 + 3][br] = bReg[u].w;                                   \
    }                                                                      \
  }

#define COMPUTE_TILE(BUF_)                                                 \
  {                                                                        \
    _Pragma("unroll")                                                      \
    for (int ks = 0; ks < KTL; ks += 4) {                                  \
      int kk = ks + (hi << 1);        /* A/B frag: VGPR0=K0/K2, VGPR1=K1/K3 */ \
      v2f a = *(const v2f*)&As[BUF_][m0 + lm][kk];                         \
      _Pragma("unroll")                                                    \
      for (int st = 0; st < 4; ++st) {                                     \
        v2f b = *(const v2f*)&Bs[BUF_][n0 + (st << 4) + lm][kk];           \
        acc[st] = __builtin_amdgcn_wmma_f32_16x16x4_f32(                   \
            false, a, false, b, (short)0, acc[st], false, false);          \
      }                                                                    \
    }                                                                      \
  }

  const int ntiles = K / KTL;
  LOAD_TILE(0);
  STORE_TILE(0);
  __syncthreads();
  for (int tI = 0; tI < ntiles; ++tI) {
    int cur = tI & 1;
    if (tI + 1 < ntiles) LOAD_TILE((tI + 1) * KTL);   // overlap with compute
    COMPUTE_TILE(cur);                                // EXEC all-ones here
    if (tI + 1 < ntiles) {
      STORE_TILE(cur ^ 1);
      __syncthreads();                                // one barrier per tile
    }
  }

  // epilogue: bias + optional relu; C/D layout: VGPR j -> rows j (lo), j+8 (hi)
#pragma unroll
  for (int st = 0; st < 4; ++st) {
    int col = blockN + n0 + (st << 4) + lm;
    if (col < N) {
      float bv = bias[col];
#pragma unroll
      for (int j = 0; j < 8; ++j) {
        int row = blockM + m0 + (hi << 3) + j;
        if (row < M) {
          float v = acc[st][j] + bv;
          if (relu) v = fmaxf(v, 0.f);
          C[(size_t)row * ldc + col] = v;
        }
      }
    }
  }
#undef LOAD_TILE
#undef STORE_TILE
#undef COMPUTE_TILE
}

// ---------------- decode (BoxCoder w=(10,10,5,5)) + softmax ----------------
__global__ __launch_bounds__(128) void decode_softmax_kernel(
    const float* __restrict__ logits, const float* __restrict__ breg,
    const float* __restrict__ proposals, float* __restrict__ scores,
    float* __restrict__ boxes)
{
  int g = blockIdx.x;
  int tid = threadIdx.x;
  __shared__ float red[128];

  float z = (tid < NCLS) ? logits[(size_t)g * NCLS + tid] : -3.0e38f;
  red[tid] = z; __syncthreads();
  for (int s = 64; s > 0; s >>= 1) {
    if (tid < s) red[tid] = fmaxf(red[tid], red[tid + s]);
    __syncthreads();
  }
  float mx = red[0]; __syncthreads();
  float e = (tid < NCLS) ? expf(z - mx) : 0.f;
  red[tid] = e; __syncthreads();
  for (int s = 64; s > 0; s >>= 1) {
    if (tid < s) red[tid] += red[tid + s];
    __syncthreads();
  }
  float denom = red[0];
  if (tid < NCLS) scores[(size_t)g * NCLS + tid] = e / denom;

  const float* p = proposals + (size_t)g * 4;
  float w  = p[2] - p[0], h  = p[3] - p[1];
  float cx = p[0] + 0.5f * w, cy = p[1] + 0.5f * h;
  if (tid < NCLS) {
    const float* d = breg + (size_t)g * (NCLS * 4) + tid * 4;
    float dx = d[0] / 10.f, dy = d[1] / 10.f;
    float dw = fminf(d[2] / 5.f, BBOX_CLIP);
    float dh = fminf(d[3] / 5.f, BBOX_CLIP);
    float pcx = dx * w + cx, pcy = dy * h + cy;
    float pw = expf(dw) * w, ph = expf(dh) * h;
    float x1 = pcx - 0.5f * pw, y1 = pcy - 0.5f * ph;
    float x2 = pcx + 0.5f * pw, y2 = pcy + 0.5f * ph;
    x1 = fminf(fmaxf(x1, 0.f), 800.f); y1 = fminf(fmaxf(y1, 0.f), 800.f);
    x2 = fminf(fmaxf(x2, 0.f), 800.f); y2 = fminf(fmaxf(y2, 0.f), 800.f);
    float* ob = boxes + ((size_t)g * NCLS + tid) * 4;
    ob[0] = x1; ob[1] = y1; ob[2] = x2; ob[3] = y2;
  }
}

// ---------------- build (key,idx) pool for per-image sort ------------------
__global__ void fill_sort_kernel(const float* __restrict__ scores,
                                 const float* __restrict__ boxes,
                                 float* __restrict__ skey, int* __restrict__ sidx)
{
  int t = blockIdx.x * blockDim.x + threadIdx.x;
  int b = t >> NSORT_LOG;
  int j = t & (NSORT - 1);
  float key = -3.4e38f;                            // pad -> sorts last
  if (j < NCAND) {
    int r = j / (NCLS - 1);
    int c = 1 + j % (NCLS - 1);                    // background dropped
    int g = b * RPROP + r;
    const float* bx = boxes + ((size_t)g * NCLS + c) * 4;
    float sc = scores[(size_t)g * NCLS + c];
    float wv = bx[2] - bx[0], hv = bx[3] - bx[1];
    bool valid = (sc > 0.05f) && (wv >= 0.01f) && (hv >= 0.01f);
    key = valid ? sc : -1.0f;
  }
  skey[t] = key;
  sidx[t] = j;
}

// comparator: descending by key, tie -> lower original index first
__device__ __forceinline__ bool kv_first(float ka, int ia, float kb, int ib) {
  return (ka > kb) || (ka == kb && ia < ib);
}

// ---------------- global bitonic pass (jj >= 2048) -------------------------
__global__ void bitonic_pass(float* __restrict__ skey, int* __restrict__ sidx,
                             int k, int jj)
{
  int t = blockIdx.x * blockDim.x + threadIdx.x;
  int b = t >> NSORT_LOG;
  int i = t & (NSORT - 1);
  int ixj = i ^ jj;
  if (ixj <= i) return;
  int base = b << NSORT_LOG;
  float ka = skey[base + i], kb = skey[base + ixj];
  int   ia = sidx[base + i], ib = sidx[base + ixj];
  bool desc = ((i & k) == 0);
  if (desc ? !kv_first(ka, ia, kb, ib) : kv_first(ka, ia, kb, ib)) {
    skey[base + i] = kb; skey[base + ixj] = ka;
    sidx[base + i] = ib; sidx[base + ixj] = ia;
  }
}

// ---------------- LDS bitonic: completes all k in [2, 2048] ----------------
__global__ __launch_bounds__(256) void bitonic_sort_local(
    float* __restrict__ skey, int* __restrict__ sidx)
{
  __shared__ float ks[2048];
  __shared__ int   is[2048];
  int gbase = blockIdx.x << 11;       // windows never cross an image boundary
#pragma unroll
  for (int u = 0; u < 8; ++u) {
    int l = threadIdx.x + (u << 8);
    ks[l] = skey[gbase + l]; is[l] = sidx[gbase + l];
  }
  __syncthreads();
  for (int k = 2; k <= 2048; k <<= 1) {
    for (int jj = k >> 1; jj > 0; jj >>= 1) {
#pragma unroll
      for (int u = 0; u < 4; ++u) {
        int p = threadIdx.x + (u << 8);
        int i = ((p & ~(jj - 1)) << 1) | (p & (jj - 1));
        int ixj = i | jj;
        int il = (gbase + i) & (NSORT - 1);       // image-local index
        bool desc = ((il & k) == 0);
        float ka = ks[i], kb = ks[ixj];
        int   ia = is[i], ib = is[ixj];
        if (desc ? !kv_first(ka, ia, kb, ib) : kv_first(ka, ia, kb, ib)) {
          ks[i] = kb; ks[ixj] = ka; is[i] = ib; is[ixj] = ia;
        }
      }
      __syncthreads();
    }
  }
#pragma unroll
  for (int u = 0; u < 8; ++u) {
    int l = threadIdx.x + (u << 8);
    skey[gbase + l] = ks[l]; sidx[gbase + l] = is[l];
  }
}

// ---------------- LDS bitonic: finishes jj in [1024, 1] for given k --------
__global__ __launch_bounds__(256) void bitonic_local_finish(
    float* __restrict__ skey, int* __restrict__ sidx, int k)
{
  __shared__ float ks[2048];
  __shared__ int   is[2048];
  int gbase = blockIdx.x << 11;
#pragma unroll
  for (int u = 0; u < 8; ++u) {
    int l = threadIdx.x + (u << 8);
    ks[l] = skey[gbase + l]; is[l] = sidx[gbase + l];
  }
  __syncthreads();
  for (int jj = 1024; jj > 0; jj >>= 1) {
#pragma unroll
    for (int u = 0; u < 4; ++u) {
      int p = threadIdx.x + (u << 8);
      int i = ((p & ~(jj - 1)) << 1) | (p & (jj - 1));
      int ixj = i | jj;
      int il = (gbase + i) & (NSORT - 1);
      bool desc = ((il & k) == 0);
      float ka = ks[i], kb = ks[ixj];
      int   ia = is[i], ib = is[ixj];
      if (desc ? !kv_first(ka, ia, kb, ib) : kv_first(ka, ia, kb, ib)) {
        ks[i] = kb; ks[ixj] = ka; is[i] = ib; is[ixj] = ia;
      }
    }
    __syncthreads();
  }
#pragma unroll
  for (int u = 0; u < 8; ++u) {
    int l = threadIdx.x + (u << 8);
    skey[gbase + l] = ks[l]; sidx[gbase + l] = is[l];
  }
}

// ---------------- per-image mask NMS + final top-100 -----------------------
__global__ __launch_bounds__(1024) void nms_kernel(
    const float* __restrict__ skey, const int* __restrict__ sidx,
    const float* __restrict__ boxes, unsigned* __restrict__ maskws,
    float* __restrict__ out)
{
  int b = blockIdx.x;
  int tid = threadIdx.x;                    // slot 0..1023 (sorted order)
  __shared__ float s_ob[KEEPN * 4];         // class-offset boxes
  __shared__ float s_bk[KEEPN * 4];         // original boxes
  __shared__ float s_sc[KEEPN];
  __shared__ float s_area[KEEPN];
  __shared__ int   s_lb[KEEPN];
  __shared__ int   s_keep[KEEPN];
  __shared__ float s_key[KEEPN];
  __shared__ int   s_fi[KEEPN];
  __shared__ unsigned s_kw[32];

  int base = b << NSORT_LOG;
  float key = skey[base + tid];
  int   j   = sidx[base + tid];

  float b0 = 0.f, b1 = 0.f, b2 = 0.f, b3 = 0.f;
  int lb = 0;
  if (key > -1.0e38f) {                     // real candidate (key >= -1.0)
    int r = j / (NCLS - 1);
    int c = 1 + j % (NCLS - 1);
    int g = b * RPROP + r;
    const float* p = boxes + ((size_t)g * NCLS + c) * 4;
    b0 = p[0]; b1 = p[1]; b2 = p[2]; b3 = p[3];
    lb = c;
  }
  bool vk = (key > 0.0f) && (tid < RPROP);  // valid implies score>0.05>0
  float off = (float)lb * 4096.0f;          // batched_nms class offset
  s_bk[tid * 4 + 0] = b0; s_bk[tid * 4 + 1] = b1;
  s_bk[tid * 4 + 2] = b2; s_bk[tid * 4 + 3] = b3;
  s_ob[tid * 4 + 0] = b0 + off; s_ob[tid * 4 + 1] = b1 + off;
  s_ob[tid * 4 + 2] = b2 + off; s_ob[tid * 4 + 3] = b3 + off;
  s_area[tid] = (b2 - b0) * (b3 - b1);
  s_sc[tid]   = (tid < RPROP) ? key : -2.0f;
  s_lb[tid]   = lb;
  s_keep[tid] = vk ? 1 : 0;
  __syncthreads();

  // phase 1: suppression mask matrix, RPROP rows x 32 words (j in [0,1024))
  unsigned* mask = maskws + (size_t)b * (RPROP * 32);
  for (int w = tid; w < RPROP * 32; w += 1024) {
    int i = w >> 5, wl = w & 31;
    float ax0 = s_ob[i * 4 + 0], ay0 = s_ob[i * 4 + 1];
    float ax1 = s_ob[i * 4 + 2], ay1 = s_ob[i * 4 + 3];
    float aa  = s_area[i];
    unsigned bits = 0u;
    int j0 = wl << 5;
#pragma unroll
    for (int e = 0; e < 32; ++e) {
      int jj2 = j0 + e;
      if (jj2 > i) {
        float lx = fmaxf(ax0, s_ob[jj2 * 4 + 0]);
        float ly = fmaxf(ay0, s_ob[jj2 * 4 + 1]);
        float rx = fminf(ax1, s_ob[jj2 * 4 + 2]);
        float ry = fminf(ay1, s_ob[jj2 * 4 + 3]);
        float iw = fmaxf(rx - lx, 0.f), ih = fmaxf(ry - ly, 0.f);
        float inter = iw * ih;
        float iou = inter / (aa + s_area[jj2] - inter + 1e-9f);
        if (iou > 0.5f) bits |= (1u << e);
      }
    }
    mask[w] = bits;
  }
  __threadfence();
  __syncthreads();

  // phase 2: one wave replays the exact greedy scan (keep word per lane)
  if (tid < 32) {
    unsigned kw = 0u;
    for (int e = 0; e < 32; ++e)
      kw |= (s_keep[(tid << 5) + e] ? 1u : 0u) << e;
    unsigned mnext = mask[tid];             // row 0, word `tid`
    for (int i = 0; i < RPROP; ++i) {
      unsigned mcur = mnext;
      if (i + 1 < RPROP) mnext = mask[((i + 1) << 5) + tid];
      unsigned kword = (unsigned)__shfl((int)kw, i >> 5, 32);
      if ((kword >> (i & 31)) & 1u) kw &= ~mcur;
    }
    s_kw[tid] = kw;
  }
  __syncthreads();

  // final top-100: in-LDS bitonic over the 1024 slots
  int keep_me = (s_kw[tid >> 5] >> (tid & 31)) & 1;
  s_key[tid] = keep_me ? s_sc[tid] : ((tid < RPROP) ? -1.0f : -2.0f);
  s_fi[tid]  = tid;
  __syncthreads();
  for (int k2 = 2; k2 <= KEEPN; k2 <<= 1) {
    for (int jj = k2 >> 1; jj > 0; jj >>= 1) {
      int ixj = tid ^ jj;
      if (ixj > tid) {
        float ka = s_key[tid], kb = s_key[ixj];
        int   ia = s_fi[tid],  ib = s_fi[ixj];
        bool desc = ((tid & k2) == 0);
        if (desc ? !kv_first(ka, ia, kb, ib) : kv_first(ka, ia, kb, ib)) {
          s_key[tid] = kb; s_key[ixj] = ka;
          s_fi[tid]  = ib; s_fi[ixj]  = ia;
        }
      }
      __syncthreads();
    }
  }

  if (tid < 100) {
    float fs = s_key[tid];
    int   fi = s_fi[tid];
    bool  m  = fs > 0.0f;
    float* ob = out + ((size_t)b * 100 + tid) * 4;            // boxes [4,100,4]
    ob[0] = m ? s_bk[fi * 4 + 0] : 0.f;
    ob[1] = m ? s_bk[fi * 4 + 1] : 0.f;
    ob[2] = m ? s_bk[fi * 4 + 2] : 0.f;
    ob[3] = m ? s_bk[fi * 4 + 3] : 0.f;
    out[BIMG * 100 * 4 + b * 100 + tid] = m ? fs : 0.f;       // scores [4,100]
    // labels are int32 in the reference; store the int32 bit pattern
    ((int*)out)[BIMG * 100 * 5 + b * 100 + tid] = m ? s_lb[fi] : -1;
  }
}

// ---------------------------------------------------------------------------
extern "C" void kernel_launch(void* const* d_in, const int* in_sizes, int n_in,
                              void* d_out, int out_size, void* d_ws, size_t ws_size,
                              hipStream_t stream) {
  (void)in_sizes; (void)n_in; (void)out_size; (void)ws_size;
  const float* box_features = (const float*)d_in[0];   // [4000, 12544]
  const float* proposals    = (const float*)d_in[1];   // [4, 1000, 4]
  const float* W6 = (const float*)d_in[2];
  const float* b6 = (const float*)d_in[3];
  const float* W7 = (const float*)d_in[4];
  const float* b7 = (const float*)d_in[5];
  const float* Wc = (const float*)d_in[6];
  const float* bc = (const float*)d_in[7];
  const float* Wb = (const float*)d_in[8];
  const float* bb = (const float*)d_in[9];
  float* out = (float*)d_out;

  // workspace layout (~52 MB)
  float* X1     = (float*)d_ws;                    // 4000*1024
  float* X2     = X1 + (size_t)NROWS * REPD;       // 4000*1024
  float* logits = X2 + (size_t)NROWS * REPD;       // 4000*91
  float* breg   = logits + (size_t)NROWS * NCLS;   // 4000*364
  float* scores = breg + (size_t)NROWS * NCLS * 4; // 4000*91
  float* boxes  = scores + (size_t)NROWS * NCLS;   // 4000*91*4
  float* skey   = boxes + (size_t)NROWS * NCLS * 4;// 4*131072
  int*   sidx   = (int*)(skey + (size_t)BIMG * NSORT);
  unsigned* maskws = (unsigned*)(sidx + (size_t)BIMG * NSORT); // 4*32000

  dim3 blk(256);
  // FC6 + relu : 102.8 GFLOP fp32 via v_wmma_f32_16x16x4_f32
  gemm_wmma_f32<<<dim3(REPD / NT, (NROWS + MT - 1) / MT), blk, 0, stream>>>(
      box_features, W6, b6, X1, NROWS, REPD, POOLD, REPD, 1);
  // FC7 + relu
  gemm_wmma_f32<<<dim3(REPD / NT, (NROWS + MT - 1) / MT), blk, 0, stream>>>(
      X1, W7, b7, X2, NROWS, REPD, REPD, REPD, 1);
  // class logits [4000, 91]
  gemm_wmma_f32<<<dim3((NCLS + NT - 1) / NT, (NROWS + MT - 1) / MT), blk, 0, stream>>>(
      X2, Wc, bc, logits, NROWS, NCLS, REPD, NCLS, 0);
  // box regression [4000, 364]
  gemm_wmma_f32<<<dim3((NCLS * 4 + NT - 1) / NT, (NROWS + MT - 1) / MT), blk, 0, stream>>>(
      X2, Wb, bb, breg, NROWS, NCLS * 4, REPD, NCLS * 4, 0);

  // decode + softmax, one block per RoI
  decode_softmax_kernel<<<NROWS, 128, 0, stream>>>(logits, breg, proposals,
                                                   scores, boxes);

  // candidate pool + exact descending bitonic sort per image (28 launches)
  int sortThreads = BIMG * NSORT;                  // 524288
  fill_sort_kernel<<<sortThreads / 256, 256, 0, stream>>>(scores, boxes, skey, sidx);
  int nWin = sortThreads / 2048;                   // 256 windows of 2048
  bitonic_sort_local<<<nWin, 256, 0, stream>>>(skey, sidx);     // k <= 2048
  for (int k = 4096; k <= NSORT; k <<= 1) {
    for (int jj = k >> 1; jj >= 2048; jj >>= 1)
      bitonic_pass<<<sortThreads / 256, 256, 0, stream>>>(skey, sidx, k, jj);
    bitonic_local_finish<<<nWin, 256, 0, stream>>>(skey, sidx, k);
  }

  // mask-matrix greedy NMS + final top-100 per image
  nms_kernel<<<BIMG, 1024, 0, stream>>>(skey, sidx, boxes, maskws, out);
}